// BundleAdjustment_5901285064866
// MI455X (gfx1250) — compile-verified
//
#include <hip/hip_runtime.h>
#include <math.h>

#define NPOSES 4096
#define POSE_F 7   // tx ty tz qx qy qz qw

typedef int v4i __attribute__((ext_vector_type(4)));
typedef int v8i __attribute__((ext_vector_type(8)));

struct F3 { float x, y, z; };

__device__ __forceinline__ F3 f3(float x, float y, float z) { F3 r; r.x=x; r.y=y; r.z=z; return r; }
__device__ __forceinline__ F3 f3add(F3 a, F3 b) { return f3(a.x+b.x, a.y+b.y, a.z+b.z); }
__device__ __forceinline__ F3 f3sub(F3 a, F3 b) { return f3(a.x-b.x, a.y-b.y, a.z-b.z); }
__device__ __forceinline__ F3 f3scale(float s, F3 a) { return f3(s*a.x, s*a.y, s*a.z); }
__device__ __forceinline__ F3 f3cross(F3 a, F3 b) {
    return f3(a.y*b.z - a.z*b.y, a.z*b.x - a.x*b.z, a.x*b.y - a.y*b.x);
}

// rotate v by quaternion (qx,qy,qz,qw):  v + w*t + qv x t,  t = 2*(qv x v)
__device__ __forceinline__ F3 quat_rot(float qx, float qy, float qz, float qw, F3 v) {
    F3 qv = f3(qx, qy, qz);
    F3 t  = f3scale(2.0f, f3cross(qv, v));
    return f3add(f3add(v, f3scale(qw, t)), f3cross(qv, t));
}

// ---------------------------------------------------------------------------
// Kernel 1: projection residuals (2 per element) + elevation residuals.
// Stages the 4096x7 pose table into LDS with ONE Tensor Data Mover descriptor
// (tensor_load_to_lds, TENSORcnt) issued by wave 0.
// ---------------------------------------------------------------------------
__global__ __launch_bounds__(256) void residual_kernel(
    const float* __restrict__ poses,          // NPOSES*7
    const float* __restrict__ patch_coords,   // E*2  (r, theta)
    const float* __restrict__ elev,           // E
    const float* __restrict__ init_elev,      // E
    const float* __restrict__ target_coords,  // ES*2
    const int*   __restrict__ src_idx,        // ES
    const int*   __restrict__ tgt_idx,        // ES
    const int*   __restrict__ patch_idx,      // ES
    float* __restrict__ out_proj,             // ES*2
    float* __restrict__ out_elev,             // E
    int n)
{
    __shared__ float s_poses[NPOSES * POSE_F];   // 114,688 bytes (< 320KB WGP LDS)

    // ---- TDM stage of pose table into LDS (TENSORcnt path, gfx1250) ----
    if ((threadIdx.x >> 5) == 0) {               // wave 0 only (EXEC ignored by TDM)
        const unsigned NWORDS = NPOSES * POSE_F; // 28672 dwords = 114,688 B
        unsigned lds_addr = (unsigned)(size_t)(&s_poses[0]);
        unsigned long long ga = (unsigned long long)(const void*)poses;

        // D# group 0: count=1, lds_addr, global_addr[56:0], type=2
        v4i g0;
        g0.x = 1;                                         // count=1, is_restore=0
        g0.y = (int)lds_addr;                             // lds_addr[31:0]
        g0.z = (int)(unsigned)(ga & 0xffffffffu);         // global_addr[31:0]
        g0.w = (int)((unsigned)((ga >> 32) & 0x01ffffffu) // global_addr[56:32]
                     | 0x80000000u);                      // type=2 (bits 127:126)

        // D# group 1: data_size=4B, 1 row of 28672 elems, stride 28672, no pad
        v8i g1;
        g1[0] = 0x20000;                  // workgroup_mask=0, data_size=2 (4B)
        g1[1] = (int)(NWORDS << 16);      // tensor_dim0[15:0] @ bits 63:48
        g1[2] = (int)(1u << 16);          // tensor_dim0[31:16]=0, tensor_dim1=1
        g1[3] = (int)(NWORDS << 16);      // tensor_dim1[31:16]=0, tile_dim0=28672
        g1[4] = 1;                        // tile_dim1=1, tile_dim2=0
        g1[5] = (int)NWORDS;              // tensor_dim0_stride[31:0]
        g1[6] = (int)(NWORDS << 16);      // stride0[47:32]=0, dim1_stride[15:0]
        g1[7] = 0;                        // dim1_stride[47:16]=0

        asm volatile("tensor_load_to_lds %0, %1"
                     :: "s"(g0), "s"(g1)
                     : "memory");
        asm volatile("s_wait_tensorcnt 0x0" ::: "memory");
    }
    __syncthreads();

    const int stride = (int)(gridDim.x * blockDim.x);
    for (int e = (int)(blockIdx.x * blockDim.x + threadIdx.x); e < n; e += stride) {
        int si = src_idx[e];
        int ti = tgt_idx[e];
        int pi = patch_idx[e];

        // polar -> cart
        float r  = patch_coords[2 * pi + 0];
        float th = patch_coords[2 * pi + 1];
        float ph = elev[pi];
        float sth, cth, sph, cph;
        __sincosf(th, &sth, &cth);
        __sincosf(ph, &sph, &cph);
        F3 cart = f3(r * cph * cth, r * cph * sth, r * sph);

        // gather source / target poses from LDS
        int sb = si * POSE_F;
        float stx = s_poses[sb+0], sty = s_poses[sb+1], stz = s_poses[sb+2];
        float sqx = s_poses[sb+3], sqy = s_poses[sb+4], sqz = s_poses[sb+5], sqw = s_poses[sb+6];
        int tb = ti * POSE_F;
        float ttx = s_poses[tb+0], tty = s_poses[tb+1], ttz = s_poses[tb+2];
        float tqx = s_poses[tb+3], tqy = s_poses[tb+4], tqz = s_poses[tb+5], tqw = s_poses[tb+6];

        // g = R_src * cart + t_src
        F3 g = f3add(quat_rot(sqx, sqy, sqz, sqw, cart), f3(stx, sty, stz));
        // loc = R_tgt^T * (g - t_tgt)   (se3_inv then act == rotate-by-conjugate of diff)
        F3 d   = f3sub(g, f3(ttx, tty, ttz));
        F3 loc = quat_rot(-tqx, -tqy, -tqz, tqw, d);

        // cart -> polar (only first two components used)
        float rr = sqrtf(loc.x*loc.x + loc.y*loc.y + loc.z*loc.z + 1e-24f);
        float az = atan2f(loc.y, loc.x);

        out_proj[2*e + 0] = rr - target_coords[2*e + 0];
        out_proj[2*e + 1] = az - target_coords[2*e + 1];

        // elevation residual (linear index, not patch_idx)
        out_elev[e] = elev[e] - init_elev[e];
    }
}

// ---------------------------------------------------------------------------
// Kernel 2: pose residuals  se3_log( se3_inv(init_poses) * poses )  (6 per pose)
// ---------------------------------------------------------------------------
__global__ __launch_bounds__(256) void pose_res_kernel(
    const float* __restrict__ poses,
    const float* __restrict__ init_poses,
    float* __restrict__ out,   // P*6
    int n)
{
    int p = (int)(blockIdx.x * blockDim.x + threadIdx.x);
    if (p >= n) return;

    const float* ip = init_poses + (size_t)p * POSE_F;
    const float* bp = poses      + (size_t)p * POSE_F;
    float itx = ip[0], ity = ip[1], itz = ip[2];
    float iqx = ip[3], iqy = ip[4], iqz = ip[5], iqw = ip[6];
    float btx = bp[0], bty = bp[1], btz = bp[2];
    float bqx = bp[3], bqy = bp[4], bqz = bp[5], bqw = bp[6];

    // A = inv(init):  Aq = conj(iq),  At = -R(Aq)*it
    float aqx = -iqx, aqy = -iqy, aqz = -iqz, aqw = iqw;
    F3 At = f3scale(-1.0f, quat_rot(aqx, aqy, aqz, aqw, f3(itx, ity, itz)));

    // T = A * B :  q = Aq*Bq,  t = R(Aq)*Bt + At
    float qx = aqw*bqx + aqx*bqw + aqy*bqz - aqz*bqy;
    float qy = aqw*bqy - aqx*bqz + aqy*bqw + aqz*bqx;
    float qz = aqw*bqz + aqx*bqy - aqy*bqx + aqz*bqw;
    float qw = aqw*bqw - aqx*bqx - aqy*bqy - aqz*bqz;
    F3 t = f3add(quat_rot(aqx, aqy, aqz, aqw, f3(btx, bty, btz)), At);

    // so3_log(q)
    if (qw < 0.0f) { qx = -qx; qy = -qy; qz = -qz; qw = -qw; }
    float nn = sqrtf(qx*qx + qy*qy + qz*qz + 1e-24f);
    float theta = 2.0f * atan2f(nn, qw);
    bool small_n = nn < 1e-7f;
    float factor = small_n ? (2.0f / fmaxf(qw, 1e-7f)) : (theta / nn);
    F3 w = f3(factor * qx, factor * qy, factor * qz);

    // se3_log
    float th2  = w.x*w.x + w.y*w.y + w.z*w.z;
    float thv  = sqrtf(th2 + 1e-24f);
    float half = 0.5f * thv;
    float sh, ch;
    __sincosf(half, &sh, &ch);
    float coef = (thv < 1e-5f)
                   ? (1.0f / 12.0f)
                   : ((1.0f - half * ch / fmaxf(sh, 1e-12f)) / fmaxf(th2, 1e-24f));
    F3 wxt = f3cross(w, t);
    F3 tau = f3add(f3sub(t, f3scale(0.5f, wxt)), f3scale(coef, f3cross(w, wxt)));

    float* o = out + (size_t)p * 6;
    o[0] = tau.x; o[1] = tau.y; o[2] = tau.z;
    o[3] = w.x;   o[4] = w.y;   o[5] = w.z;
}

// ---------------------------------------------------------------------------
extern "C" void kernel_launch(void* const* d_in, const int* in_sizes, int n_in,
                              void* d_out, int out_size, void* d_ws, size_t ws_size,
                              hipStream_t stream)
{
    const float* poses         = (const float*)d_in[0];
    const float* patch_coords  = (const float*)d_in[1];
    const float* elev          = (const float*)d_in[2];
    const float* init_poses    = (const float*)d_in[3];
    const float* init_elev     = (const float*)d_in[4];
    const float* target_coords = (const float*)d_in[5];
    const int*   src_idx       = (const int*)d_in[6];
    const int*   tgt_idx       = (const int*)d_in[7];
    const int*   patch_idx     = (const int*)d_in[8];
    float* out = (float*)d_out;

    const int P  = in_sizes[0] / POSE_F;   // 4096
    const int ES = in_sizes[6];            // 1,048,576

    float* out_proj = out;                                // 2*ES
    float* out_pose = out + (size_t)2 * ES;               // 6*P
    float* out_elev = out_pose + (size_t)6 * P;           // E (== ES)

    residual_kernel<<<1024, 256, 0, stream>>>(
        poses, patch_coords, elev, init_elev, target_coords,
        src_idx, tgt_idx, patch_idx, out_proj, out_elev, ES);

    pose_res_kernel<<<(P + 255) / 256, 256, 0, stream>>>(
        poses, init_poses, out_pose, P);
}